// FusedMHC_84765474554153
// MI455X (gfx1250) — compile-verified
//
#include <hip/hip_runtime.h>
#include <cstdint>

// ---------------------------------------------------------------------------
// Sinkhorn-Knopp on 2M independent 8x8 matrices, f32, 5 iterations.
// Memory-bound: 1 GiB traffic -> ~46 us floor at 23.3 TB/s.
// Strategy (MI455X / gfx1250, wave32):
//   * async global->LDS streaming copies (ASYNCcnt) for perfect coalescing
//   * one matrix per lane in VGPRs -> zero cross-lane ops in the solver
//   * scaling-vector Sinkhorn:  c=1; 5x { r = 1/(M c); c = 1/(M^T r) };
//     out = diag(r) M diag(c).  Algebraically identical to the reference's
//     alternating row/col normalization, but M is never rewritten and every
//     dot product is an 8-FMA chain  -> ~25% fewer VALU issue slots.
//   * LDS padded to 272 B/matrix -> only 2-way bank conflict on ds_load_b128
//   * per-wave private LDS regions -> no barriers, only s_wait_asynccnt/dscnt
//   * v_exp_f32 + v_rcp_f32(+1 Newton, ~0.5 ulp) for exp / reciprocals
// ---------------------------------------------------------------------------

#define WAVES_PER_BLOCK   4
#define THREADS_PER_BLOCK 128
#define MAT_BYTES         256           // 8*8*4
#define MAT_STRIDE_LDS    272           // 256 + 16 pad (bank-conflict dodge)
#define WAVE_MATS         32            // one matrix per lane
#define WAVE_LDS_BYTES    (WAVE_MATS * MAT_STRIDE_LDS)   // 8704
#define BLOCK_MATS        (WAVES_PER_BLOCK * WAVE_MATS)  // 128
#define WAVE_GBYTES       (WAVE_MATS * MAT_BYTES)        // 8192

__device__ __forceinline__ float rcp_nr(float s) {
    float t = __builtin_amdgcn_rcpf(s);
    return t * (2.0f - s * t);          // one Newton step -> ~0.5 ulp
}

__launch_bounds__(THREADS_PER_BLOCK)
__global__ void sinkhorn8x8_kernel(const float* __restrict__ in,
                                   float* __restrict__ out,
                                   int nmat)
{
    __shared__ __align__(16) unsigned char lds[WAVES_PER_BLOCK * WAVE_LDS_BYTES];

    const int lane = threadIdx.x & 31;          // wave32 on gfx1250
    const int wave = threadIdx.x >> 5;

    const long long blockMat = (long long)blockIdx.x * BLOCK_MATS;
    if (blockMat >= nmat) return;               // grid sized to full blocks

    // Low 32 bits of a generic pointer to LDS == byte offset in LDS aperture.
    const uint32_t ldsBase = (uint32_t)(uintptr_t)&lds[0];
    const uint32_t waveLds = ldsBase + (uint32_t)wave * WAVE_LDS_BYTES;

    // Uniform 64-bit bases for the block (SADDR of the async ops).
    const unsigned char* gin  = (const unsigned char*)in  + blockMat * MAT_BYTES;
    unsigned char*       gout = (unsigned char*)out       + blockMat * MAT_BYTES;

    // ---------------- async load: global (contiguous) -> LDS (padded) ------
    // step k copies matrices 2k (lanes 0-15) and 2k+1 (lanes 16-31):
    //   global granule (k*32 + l) at byte (k*32+l)*16  (512 B / instruction)
    //   lds  addr = waveLds + (2k + l>>4)*272 + (l&15)*16
    {
        uint32_t ldsA = waveLds
                      + (uint32_t)(lane >> 4) * MAT_STRIDE_LDS
                      + (uint32_t)(lane & 15) * 16u;
        uint32_t gOff = (uint32_t)wave * WAVE_GBYTES + (uint32_t)lane * 16u;
        #pragma unroll
        for (int k = 0; k < 16; ++k) {
            asm volatile("global_load_async_to_lds_b128 %0, %1, %2"
                         :: "v"(ldsA), "v"(gOff), "s"(gin) : "memory");
            ldsA += 2u * MAT_STRIDE_LDS;   // 544: padded LDS stride
            gOff += 512u;                  // dense global stride
        }
        asm volatile("s_wait_asynccnt 0x0" ::: "memory");
    }

    // ---------------- LDS -> VGPRs: this lane's whole 8x8 matrix -----------
    float m[64];
    {
        const unsigned char* basep =
            lds + (size_t)wave * WAVE_LDS_BYTES + (size_t)lane * MAT_STRIDE_LDS;
        #pragma unroll
        for (int g = 0; g < 16; ++g) {
            float4 v = *(const float4*)(basep + g * 16);
            m[4 * g + 0] = v.x; m[4 * g + 1] = v.y;
            m[4 * g + 2] = v.z; m[4 * g + 3] = v.w;
        }
    }

    // ---------------- M = exp(x)  (v_exp_f32 path) -------------------------
    #pragma unroll
    for (int i = 0; i < 64; ++i)
        m[i] = __builtin_amdgcn_exp2f(m[i] * 1.44269504088896340736f);

    // ---------------- Sinkhorn via scaling vectors -------------------------
    float r[8], c[8];

    // iteration 1: c == 1  ->  r = 1/rowsum(M)  (pure adds)
    #pragma unroll
    for (int i = 0; i < 8; ++i) {
        float s = m[8 * i + 0];
        #pragma unroll
        for (int j = 1; j < 8; ++j) s += m[8 * i + j];
        r[i] = rcp_nr(s);
    }
    #pragma unroll
    for (int j = 0; j < 8; ++j) {           // c = 1/(M^T r)
        float s = m[j] * r[0];
        #pragma unroll
        for (int i = 1; i < 8; ++i) s += m[8 * i + j] * r[i];  // v_fma chains
        c[j] = rcp_nr(s);
    }

    // iterations 2..5
    #pragma unroll
    for (int it = 0; it < 4; ++it) {
        #pragma unroll
        for (int i = 0; i < 8; ++i) {       // r = 1/(M c)
            float s = m[8 * i + 0] * c[0];
            #pragma unroll
            for (int j = 1; j < 8; ++j) s += m[8 * i + j] * c[j];
            r[i] = rcp_nr(s);
        }
        #pragma unroll
        for (int j = 0; j < 8; ++j) {       // c = 1/(M^T r)
            float s = m[j] * r[0];
            #pragma unroll
            for (int i = 1; i < 8; ++i) s += m[8 * i + j] * r[i];
            c[j] = rcp_nr(s);
        }
    }

    // ---------------- out = diag(r) M diag(c) -> LDS (padded layout) -------
    {
        unsigned char* basep =
            lds + (size_t)wave * WAVE_LDS_BYTES + (size_t)lane * MAT_STRIDE_LDS;
        #pragma unroll
        for (int g = 0; g < 16; ++g) {
            const int row = g >> 1;          // 2 granules (8 floats) per row
            const int jb  = (g & 1) * 4;     // column base within the row
            const float rr = r[row];
            float4 v;
            v.x = m[8 * row + jb + 0] * rr * c[jb + 0];
            v.y = m[8 * row + jb + 1] * rr * c[jb + 1];
            v.z = m[8 * row + jb + 2] * rr * c[jb + 2];
            v.w = m[8 * row + jb + 3] * rr * c[jb + 3];
            *(float4*)(basep + g * 16) = v;
        }
    }
    // DS stores must land in LDS before the async engine reads them.
    asm volatile("s_wait_dscnt 0x0" ::: "memory");

    // ---------------- async store: LDS (padded) -> global (contiguous) -----
    {
        uint32_t ldsA = waveLds
                      + (uint32_t)(lane >> 4) * MAT_STRIDE_LDS
                      + (uint32_t)(lane & 15) * 16u;
        uint32_t gOff = (uint32_t)wave * WAVE_GBYTES + (uint32_t)lane * 16u;
        #pragma unroll
        for (int k = 0; k < 16; ++k) {
            asm volatile("global_store_async_from_lds_b128 %0, %1, %2"
                         :: "v"(gOff), "v"(ldsA), "s"(gout) : "memory");
            ldsA += 2u * MAT_STRIDE_LDS;
            gOff += 512u;
        }
        asm volatile("s_wait_asynccnt 0x0" ::: "memory");
    }
}

extern "C" void kernel_launch(void* const* d_in, const int* in_sizes, int n_in,
                              void* d_out, int out_size, void* d_ws, size_t ws_size,
                              hipStream_t stream)
{
    const float* x   = (const float*)d_in[0];
    float*       outp = (float*)d_out;

    const long long nelem = (long long)in_sizes[0];   // 2048*1024*8*8
    const long long nmat  = nelem / 64;               // 2,097,152 (divisible by 128)
    const int       grid  = (int)(nmat / BLOCK_MATS); // 16384 full blocks

    sinkhorn8x8_kernel<<<grid, THREADS_PER_BLOCK, 0, stream>>>(x, outp, (int)nmat);
}